// GATv2Layer_39479339385232
// MI455X (gfx1250) — compile-verified
//
#include <hip/hip_runtime.h>
#include <math.h>

#define N_NODES 50000
#define N_EDGES 800000
#define D_IN    128
#define D_E     64
#define NH      128   // HEADS * D_OUT
#define HEADS   4
#define NEG_SLOPE 0.2f
#define EPS 1e-16f

typedef float v2f __attribute__((ext_vector_type(2)));
typedef float v8f __attribute__((ext_vector_type(8)));

// ---- ordered-uint encoding so atomicMax(uint) == float max (handles negatives) ----
__device__ __forceinline__ unsigned f2ord(float f) {
    unsigned u = __float_as_uint(f);
    return (u & 0x80000000u) ? ~u : (u | 0x80000000u);
}
__device__ __forceinline__ float ord2f(unsigned u) {
    return (u & 0x80000000u) ? __uint_as_float(u & 0x7FFFFFFFu) : __uint_as_float(~u);
}
#define ORD_NEG_INF 0x007FFFFFu  // f2ord(-inf)

// ---------------------------------------------------------------------------
// K0: zero output, init segment-max to -inf, segment-sum to 0
// ---------------------------------------------------------------------------
__global__ __launch_bounds__(256) void k_init(float* __restrict__ out,
                                              unsigned* __restrict__ a_max,
                                              float* __restrict__ a_sum) {
    int i = blockIdx.x * blockDim.x + threadIdx.x;
    if (i < N_NODES * NH) out[i] = 0.0f;
    if (i < N_NODES * HEADS) { a_max[i] = ORD_NEG_INF; a_sum[i] = 0.0f; }
}

// ---------------------------------------------------------------------------
// K1: HT[50000,128] = H[50000,128] @ W^T   (W is [128,128] row-major, torch layout)
// One block = 16-row M tile; 8 waves = 8 N tiles of 16. f32 WMMA 16x16x4, 32 k-steps.
// ---------------------------------------------------------------------------
__global__ __launch_bounds__(256) void k_node_gemm(const float* __restrict__ H,
                                                   const float* __restrict__ W,
                                                   float* __restrict__ HT) {
    __shared__ float At[16][130];   // 130-float stride => 2-bank row skew, no ds conflicts
    const int tid = threadIdx.x;
    const int m0  = blockIdx.x * 16;

    for (int i = tid; i < 16 * 128; i += 256) {
        int r = i >> 7, cc = i & 127;
        At[r][cc] = H[(size_t)(m0 + r) * D_IN + cc];
    }
    __syncthreads();

    const int wv  = tid >> 5;
    const int lane = tid & 31;
    const int hf  = lane >> 4;      // half-wave: K-pair select
    const int l16 = lane & 15;
    const int n0  = wv * 16;

    v8f c = {0.f, 0.f, 0.f, 0.f, 0.f, 0.f, 0.f, 0.f};
    for (int k = 0; k < 128; k += 4) {
        const int kb = k + hf * 2;
        // A 16x4: lanes 0-15 rows M=0..15 K={k,k+1}; lanes 16-31 K={k+2,k+3}
        v2f a = *(const v2f*)&At[l16][kb];
        // B 4x16: B[k][n] = W[n][k]; lanes hold columns n, same K split
        v2f b = *(const v2f*)&W[(size_t)(n0 + l16) * D_IN + kb];
        c = __builtin_amdgcn_wmma_f32_16x16x4_f32(false, a, false, b,
                                                  (short)0, c, false, false);
    }
    // C/D: VGPR v -> row v (lanes 0-15) or v+8 (lanes 16-31), col = n0+l16
#pragma unroll
    for (int v = 0; v < 8; ++v)
        HT[(size_t)(m0 + v + 8 * hf) * NH + n0 + l16] = c[v];
}

// ---------------------------------------------------------------------------
// K2 (fused): for 16 edges, compute ET tile = E_tile @ W_e^T in registers via WMMA,
// add gathered HT[src]+HT[dst], leaky-relu, multiply att, reduce over d_out ->
// alpha_raw[e][h]; atomic segment-max into a_max[dst][h]. ET never hits memory.
// ---------------------------------------------------------------------------
__global__ __launch_bounds__(256) void k_edge_alpha(const float* __restrict__ E,
                                                    const float* __restrict__ We,
                                                    const float* __restrict__ HT,
                                                    const float* __restrict__ att,
                                                    const int* __restrict__ ei,
                                                    float* __restrict__ alpha,
                                                    unsigned* __restrict__ a_max) {
    __shared__ float Et[16][66];          // 16 edges x 64 (skewed)
    __shared__ int   s_src[16], s_dst[16];
    __shared__ float part[8][16];         // per-wave partial head sums per edge-row
    const int tid = threadIdx.x;
    const int e0  = blockIdx.x * 16;

    for (int i = tid; i < 16 * 64; i += 256) {
        int r = i >> 6, cc = i & 63;
        Et[r][cc] = E[(size_t)(e0 + r) * D_E + cc];
    }
    if (tid < 16) {
        s_src[tid] = ei[e0 + tid];
        s_dst[tid] = ei[N_EDGES + e0 + tid];
    }
    __syncthreads();

    const int wv  = tid >> 5;
    const int lane = tid & 31;
    const int hf  = lane >> 4;
    const int l16 = lane & 15;
    const int n0  = wv * 16;
    const int n   = n0 + l16;             // output feature column 0..127

    v8f c = {0.f, 0.f, 0.f, 0.f, 0.f, 0.f, 0.f, 0.f};
    for (int k = 0; k < 64; k += 4) {
        const int kb = k + hf * 2;
        v2f a = *(const v2f*)&Et[l16][kb];
        v2f b = *(const v2f*)&We[(size_t)n * D_E + kb];
        c = __builtin_amdgcn_wmma_f32_16x16x4_f32(false, a, false, b,
                                                  (short)0, c, false, false);
    }

    const float av = att[n];              // att flat [heads*d_out] == column index
#pragma unroll
    for (int v = 0; v < 8; ++v) {
        const int m = v + 8 * hf;         // edge row within tile
        const int s = s_src[m], d = s_dst[m];
        float x = c[v] + HT[(size_t)s * NH + n] + HT[(size_t)d * NH + n];
        x = (x > 0.f) ? x : NEG_SLOPE * x;
        x *= av;
        // reduce across the 16 lanes of this half-wave (16 d_out columns)
        x += __shfl_xor(x, 1, 32);
        x += __shfl_xor(x, 2, 32);
        x += __shfl_xor(x, 4, 32);
        x += __shfl_xor(x, 8, 32);
        if (l16 == 0) part[wv][m] = x;
    }
    __syncthreads();

    // head h spans two waves (2x16 columns): combine, store raw score, seg-max
    if (tid < 64) {
        const int el = tid & 15, h = tid >> 4;
        const float a2 = part[2 * h][el] + part[2 * h + 1][el];
        alpha[(size_t)(e0 + el) * HEADS + h] = a2;
        atomicMax(&a_max[(size_t)s_dst[el] * HEADS + h], f2ord(a2));
    }
}

// ---------------------------------------------------------------------------
// K3: w = exp(alpha - max[dst]); in-place; segment-sum via hw f32 atomics
// ---------------------------------------------------------------------------
__global__ __launch_bounds__(256) void k_softmax(const int* __restrict__ ei,
                                                 const unsigned* __restrict__ a_max,
                                                 float* __restrict__ alpha,
                                                 float* __restrict__ a_sum) {
    int i = blockIdx.x * blockDim.x + threadIdx.x;
    if (i >= N_EDGES * HEADS) return;
    const int e = i >> 2, h = i & 3;
    const int d = ei[N_EDGES + e];
    const float m = ord2f(a_max[(size_t)d * HEADS + h]);
    const float w = expf(alpha[i] - m);
    alpha[i] = w;
    unsafeAtomicAdd(&a_sum[(size_t)d * HEADS + h], w);
}

// ---------------------------------------------------------------------------
// K4: coeff = w/(sum+eps); out[dst] += HT[src]*coeff  (float4 per thread, hw atomics)
// ---------------------------------------------------------------------------
__global__ __launch_bounds__(256) void k_aggregate(const int* __restrict__ ei,
                                                   const float* __restrict__ alpha,
                                                   const float* __restrict__ a_sum,
                                                   const float* __restrict__ HT,
                                                   float* __restrict__ out) {
    int i = blockIdx.x * blockDim.x + threadIdx.x;
    if (i >= N_EDGES * 32) return;               // 32 quads of 4 floats per edge
    const int e = i >> 5;
    const int n = (i & 31) * 4;                  // feature column 0..124
    const int h = n >> 5;                        // head = n / 32
    const int s = ei[e], d = ei[N_EDGES + e];
    const float w   = alpha[(size_t)e * HEADS + h];
    const float sum = a_sum[(size_t)d * HEADS + h];
    const float coeff = w / (sum + EPS);
    const float4 ht = *(const float4*)&HT[(size_t)s * NH + n];
    float* ob = &out[(size_t)d * NH + n];
    unsafeAtomicAdd(ob + 0, ht.x * coeff);
    unsafeAtomicAdd(ob + 1, ht.y * coeff);
    unsafeAtomicAdd(ob + 2, ht.z * coeff);
    unsafeAtomicAdd(ob + 3, ht.w * coeff);
}

// ---------------------------------------------------------------------------
extern "C" void kernel_launch(void* const* d_in, const int* in_sizes, int n_in,
                              void* d_out, int out_size, void* d_ws, size_t ws_size,
                              hipStream_t stream) {
    const float* H   = (const float*)d_in[0];
    const int*   ei  = (const int*)d_in[1];
    const float* E   = (const float*)d_in[2];
    const float* W   = (const float*)d_in[3];
    const float* We  = (const float*)d_in[4];
    const float* att = (const float*)d_in[5];
    float* out = (float*)d_out;

    // workspace layout (~40 MB)
    float*    HT    = (float*)d_ws;                                  // 6,400,000 f
    float*    alpha = HT + (size_t)N_NODES * NH;                     // 3,200,000 f
    unsigned* amax  = (unsigned*)(alpha + (size_t)N_EDGES * HEADS);  //   200,000 u
    float*    asum  = (float*)(amax + (size_t)N_NODES * HEADS);      //   200,000 f

    k_init<<<(N_NODES * NH + 255) / 256, 256, 0, stream>>>(out, amax, asum);
    k_node_gemm<<<N_NODES / 16, 256, 0, stream>>>(H, W, HT);
    k_edge_alpha<<<N_EDGES / 16, 256, 0, stream>>>(E, We, HT, att, ei, alpha, amax);
    k_softmax<<<(N_EDGES * HEADS + 255) / 256, 256, 0, stream>>>(ei, amax, alpha, asum);
    k_aggregate<<<(N_EDGES * 32 + 255) / 256, 256, 0, stream>>>(ei, alpha, asum, HT, out);
}